// DiscreteHazardLoss_34892314313021
// MI455X (gfx1250) — compile-verified
//
#include <hip/hip_runtime.h>

// DiscreteHazardLoss on MI455X (gfx1250, wave32).
//
// Roofline: ~280 MB in / 4 B out @ 23.3 TB/s => ~12 us, memory bound.
// One wave handles a 16-row x 32-col tile. sum_{j<t} log(1-sigmoid(x)+eps)
// is diag(L * mask^T), computed with eight V_WMMA_F32_16X16X4_F32
// accumulations (K=4/step, 8 steps = 32 cols). Each logit is loaded exactly
// once (aligned float2 per lane per step, all loads hoisted ahead of the
// transcendental pipeline). Math path per element is lean:
//   1-sigmoid(x) = rcp(1+exp2(x*log2e));  log(u) = ln2*log2(u)
// i.e. v_exp_f32 + v_rcp_f32 + v_log_f32 + 2 mul + 2 add (no IEEE divide).
// Deterministic two-pass reduction: per-block partials in d_ws, then a
// single-block reduce writes -sum/B.

typedef __attribute__((ext_vector_type(2))) float v2f;
typedef __attribute__((ext_vector_type(8))) float v8f;

#define HZ_EPS  1e-7f
#define HZ_T    32
#define HZ_L2E  1.4426950408889634f   // log2(e)
#define HZ_LN2  0.6931471805599453f   // ln(2)
#define HZ_BLOCKS 2048
#define HZ_THREADS 256
#define HZ_WAVES_PER_BLOCK (HZ_THREADS / 32)

__device__ __forceinline__ float hz_exp(float x) {
    // e^x via v_exp_f32 (2^u); saturates to 0 / +inf out of range, which is
    // exactly the behavior the surrounding rcp/log pipeline needs.
    return __builtin_amdgcn_exp2f(x * HZ_L2E);
}

__device__ __forceinline__ float hz_log_eps(float u) {
    // ln(u + EPS), u in [0,1]; u+EPS >= 1e-7 is a normal f32, so raw
    // v_log_f32 (log2) + one multiply by ln2 is safe and cheap.
    return HZ_LN2 * __builtin_amdgcn_logf(u + HZ_EPS);
}

__global__ __launch_bounds__(HZ_THREADS)
void hazard_main_kernel(const float* __restrict__ logits,
                        const long long* __restrict__ time_bins,
                        const int* __restrict__ events,
                        float* __restrict__ partials,
                        int nrows) {
    const int lane   = threadIdx.x & 31;
    const int waveIB = threadIdx.x >> 5;
    const int gwave  = blockIdx.x * HZ_WAVES_PER_BLOCK + waveIB;
    const int nwaves = gridDim.x * HZ_WAVES_PER_BLOCK;
    const int ntiles = nrows >> 4;          // 16 rows per tile

    const int r     = lane & 15;            // row-in-tile owned by this lane
    const int khalf = lane >> 4;            // 0: K={0,1}, 1: K={2,3}

    float acc = 0.0f;                       // accumulates (log_surv + term)

    for (int tile = gwave; tile < ntiles; tile += nwaves) {
        const int myrow = (tile << 4) + r;
        const float* __restrict__ rowp = logits + (size_t)myrow * HZ_T;

        // ---- issue ALL loads for this tile up front ----
        long long tb = time_bins[myrow];
        int t = (int)(tb < 0 ? 0 : (tb > (HZ_T - 1) ? (HZ_T - 1) : tb));
        const float xt = rowp[t];           // boundary-bin logit (gather)
        const int   ev = events[myrow];

        float2 xv[8];
#pragma unroll
        for (int s = 0; s < 8; ++s)
            xv[s] = *(const float2*)(rowp + 4 * s + 2 * khalf); // 8B aligned

        // ---- 8 WMMA accumulation steps over the 32 columns ----
        v8f c = {};                          // 16x16 f32 accumulator
#pragma unroll
        for (int s = 0; s < 8; ++s) {
            const int c0 = 4 * s + 2 * khalf;       // cols c0, c0+1

            // L = log(1 - sigmoid(x) + eps) = ln2*log2(rcp(1+e^x) + eps)
            float e0 = hz_exp(xv[s].x);
            float e1 = hz_exp(xv[s].y);
            float m0 = __builtin_amdgcn_rcpf(1.0f + e0);  // 1 - sigmoid(x)
            float m1 = __builtin_amdgcn_rcpf(1.0f + e1);
            v2f a, b;
            a.x = hz_log_eps(m0);
            a.y = hz_log_eps(m1);
            // B = mask^T: B[k][n] = (col < t_row_n); same lane mapping as A
            b.x = (c0     < t) ? 1.0f : 0.0f;
            b.y = (c0 + 1 < t) ? 1.0f : 0.0f;

            // D = A(16x4) x B(4x16) + C ; diag(D) = masked row sums
            c = __builtin_amdgcn_wmma_f32_16x16x4_f32(
                    false, a, false, b, (short)0, c, false, false);
        }

        // Diagonal extraction: row i -> lane i (i<8) VGPR i, or lane i+16
        // (i>=8) VGPR i-8.  sel = lane & 7 covers both cases.
        const int sel = lane & 7;
        float dv = (sel == 0) ? c[0] : (sel == 1) ? c[1] :
                   (sel == 2) ? c[2] : (sel == 3) ? c[3] :
                   (sel == 4) ? c[4] : (sel == 5) ? c[5] :
                   (sel == 6) ? c[6] :              c[7];
        if (lane < 8 || lane >= 24) acc += dv;

        // Boundary term at bin t: computed branch-free on all lanes
        // (lanes 16-31 duplicate lanes 0-15), only lanes 0-15 accumulate.
        float et = hz_exp(-xt);
        float ht = __builtin_amdgcn_rcpf(1.0f + et);     // sigmoid(xt)
        float arg = (ev == 1) ? ht : (1.0f - ht);
        float term = hz_log_eps(arg);
        if (lane < 16) acc += term;
    }

    // wave32 tree reduction
#pragma unroll
    for (int off = 16; off > 0; off >>= 1)
        acc += __shfl_xor(acc, off, 32);

    __shared__ float sred[HZ_WAVES_PER_BLOCK];
    if (lane == 0) sred[waveIB] = acc;
    __syncthreads();
    if (threadIdx.x == 0) {
        float s = 0.0f;
#pragma unroll
        for (int i = 0; i < HZ_WAVES_PER_BLOCK; ++i) s += sred[i];
        partials[blockIdx.x] = s;
    }
}

__global__ __launch_bounds__(HZ_THREADS)
void hazard_reduce_kernel(const float* __restrict__ partials, int n,
                          float inv_rows, float* __restrict__ out) {
    float s = 0.0f;
    for (int i = threadIdx.x; i < n; i += HZ_THREADS) s += partials[i];
#pragma unroll
    for (int off = 16; off > 0; off >>= 1)
        s += __shfl_xor(s, off, 32);

    __shared__ float sred[HZ_WAVES_PER_BLOCK];
    const int lane = threadIdx.x & 31, w = threadIdx.x >> 5;
    if (lane == 0) sred[w] = s;
    __syncthreads();
    if (threadIdx.x == 0) {
        float tot = 0.0f;
#pragma unroll
        for (int i = 0; i < HZ_WAVES_PER_BLOCK; ++i) tot += sred[i];
        out[0] = -tot * inv_rows;            // mean of -(log_surv + term)
    }
}

extern "C" void kernel_launch(void* const* d_in, const int* in_sizes, int n_in,
                              void* d_out, int out_size, void* d_ws, size_t ws_size,
                              hipStream_t stream) {
    const float*     logits    = (const float*)d_in[0];
    const long long* time_bins = (const long long*)d_in[1];   // int64
    const int*       events    = (const int*)d_in[2];         // int32
    const int        nrows     = in_sizes[2];                 // B

    float* partials = (float*)d_ws;                           // HZ_BLOCKS floats

    hazard_main_kernel<<<HZ_BLOCKS, HZ_THREADS, 0, stream>>>(
        logits, time_bins, events, partials, nrows);
    hazard_reduce_kernel<<<1, HZ_THREADS, 0, stream>>>(
        partials, HZ_BLOCKS, 1.0f / (float)nrows, (float*)d_out);
}